// GATBaseline_42494406427517
// MI455X (gfx1250) — compile-verified
//
#include <hip/hip_runtime.h>
#include <cstdint>
#include <cstddef>

// MI455X / gfx1250, wave32. GAT 2-layer forward.
// GEMMs use V_WMMA_F32_16X16X4_F32 (fp32-exact tensor path); edge softmax/
// aggregation is the memory-bound bulk (one wave32 per edge, float4 lanes).

typedef float v2f __attribute__((ext_vector_type(2)));
typedef float v8f __attribute__((ext_vector_type(8)));

#define HIDC 128

// ---------------------------------------------------------------- fill
__global__ void fill_u32_kernel(uint32_t* __restrict__ p, uint32_t v, long long n) {
  long long i = (long long)blockIdx.x * blockDim.x + threadIdx.x;
  if (i < n) p[i] = v;
}

// ---------------------------------------------------------------- GEMM (WMMA f32 16x16x4)
// C[M,128] = A_rows @ W ; A row i = A[idx[i]] (idx==nullptr -> identity).
// Block: 128 threads (4 waves). Block tile: 16 rows x 128 cols; each wave owns
// two 16x16 column tiles. K = 128 in steps of 4.
__global__ void gemm16_wmma(const float* __restrict__ A, const int* __restrict__ idx,
                            const float* __restrict__ W, float* __restrict__ C, int M) {
  __shared__ float As[16][132];  // padded stride: banks (r*132+k)%64 distinct over r
  const int rowBase = blockIdx.x << 4;
  const int tid = threadIdx.x;
  for (int i = tid; i < 16 * HIDC; i += 128) {
    int r = i >> 7, c = i & (HIDC - 1);
    int row = rowBase + r;
    if (row >= M) row = M - 1;           // clamp (keeps EXEC full for WMMA)
    int arow = idx ? idx[row] : row;
    As[r][c] = A[(size_t)arow * HIDC + c];
  }
  __syncthreads();

  const int wave = tid >> 5;
  const int lane = tid & 31;
  const int half = lane >> 4;            // 0: K pair {k,k+1}, 1: {k+2,k+3}
  const int l15  = lane & 15;
  const int cb0 = (wave * 2) * 16;
  const int cb1 = cb0 + 16;

  v8f acc0 = {0.f, 0.f, 0.f, 0.f, 0.f, 0.f, 0.f, 0.f};
  v8f acc1 = {0.f, 0.f, 0.f, 0.f, 0.f, 0.f, 0.f, 0.f};

  for (int k0 = 0; k0 < HIDC; k0 += 4) {
    const int ka = k0 + 2 * half;
    v2f a; a.x = As[l15][ka]; a.y = As[l15][ka + 1];              // A 16x4 layout
    v2f b0; b0.x = W[ka * HIDC + cb0 + l15]; b0.y = W[(ka + 1) * HIDC + cb0 + l15];
    v2f b1; b1.x = W[ka * HIDC + cb1 + l15]; b1.y = W[(ka + 1) * HIDC + cb1 + l15];
    acc0 = __builtin_amdgcn_wmma_f32_16x16x4_f32(false, a, false, b0, (short)0, acc0, false, false);
    acc1 = __builtin_amdgcn_wmma_f32_16x16x4_f32(false, a, false, b1, (short)0, acc1, false, false);
  }
  // C/D layout: VGPR r -> row r (lanes 0-15) / row r+8 (lanes 16-31), col = lane&15
  for (int r = 0; r < 8; ++r) {
    int row = rowBase + r + 8 * half;
    if (row < M) {
      C[(size_t)row * HIDC + cb0 + l15] = acc0[r];
      C[(size_t)row * HIDC + cb1 + l15] = acc1[r];
    }
  }
}

// ---------------------------------------------------------------- attention logits
// als[n,h] = <h[n, h*ch : (h+1)*ch], a_src[h]> ; ald likewise.
__global__ void attn_logits_kernel(const float* __restrict__ h, const float* __restrict__ asrc,
                                   const float* __restrict__ adst, float* __restrict__ als,
                                   float* __restrict__ ald, int n, int heads, int ch) {
  int i = blockIdx.x * blockDim.x + threadIdx.x;
  if (i >= n * heads) return;
  int node = i / heads;
  int hd = i - node * heads;
  const float* hp = h + (size_t)node * HIDC + hd * ch;
  const float* asp = asrc + hd * ch;
  const float* adp = adst + hd * ch;
  float s = 0.f, d = 0.f;
  for (int c = 0; c < ch; ++c) { float v = hp[c]; s += v * asp[c]; d += v * adp[c]; }
  als[i] = s; ald[i] = d;
}

// ---------------------------------------------------------------- float max via ordered uint
__device__ __forceinline__ unsigned f2key(float f) {
  unsigned u = __float_as_uint(f);
  return (u & 0x80000000u) ? ~u : (u | 0x80000000u);
}
__device__ __forceinline__ float key2f(unsigned k) {
  unsigned u = (k & 0x80000000u) ? (k ^ 0x80000000u) : ~k;
  return __uint_as_float(u);
}

// Edge list layout: src = ei[e], dst = ei[E+e]; self-loops for e >= E.
template <int HEADS_>
__global__ void edge_max_kernel(const int* __restrict__ ei, int E, int N,
                                const float* __restrict__ als, const float* __restrict__ ald,
                                unsigned* __restrict__ mkey) {
  int e = blockIdx.x * blockDim.x + threadIdx.x;
  if (e >= E + N) return;
  int s = (e < E) ? ei[e] : (e - E);
  int d = (e < E) ? ei[E + e] : (e - E);
#pragma unroll
  for (int h = 0; h < HEADS_; ++h) {
    float v = als[s * HEADS_ + h] + ald[d * HEADS_ + h];
    v = v > 0.f ? v : 0.2f * v;                     // leaky_relu(0.2)
    atomicMax(&mkey[d * HEADS_ + h], f2key(v));
  }
}

template <int HEADS_>
__global__ void edge_expsum_kernel(const int* __restrict__ ei, int E, int N,
                                   const float* __restrict__ als, const float* __restrict__ ald,
                                   const unsigned* __restrict__ mkey, float* __restrict__ ssum) {
  int e = blockIdx.x * blockDim.x + threadIdx.x;
  if (e >= E + N) return;
  int s = (e < E) ? ei[e] : (e - E);
  int d = (e < E) ? ei[E + e] : (e - E);
#pragma unroll
  for (int h = 0; h < HEADS_; ++h) {
    float v = als[s * HEADS_ + h] + ald[d * HEADS_ + h];
    v = v > 0.f ? v : 0.2f * v;
    atomicAdd(&ssum[d * HEADS_ + h], expf(v - key2f(mkey[d * HEADS_ + h])));
  }
}

// One wave32 per edge: lane l moves channels [4l, 4l+4) as a float4
// (coalesced 512B gather per wave) and does 4 global f32 atomic adds.
template <int HEADS_>
__global__ void edge_aggregate_kernel(const int* __restrict__ ei, int E, int N,
                                      const float* __restrict__ als, const float* __restrict__ ald,
                                      const unsigned* __restrict__ mkey, const float* __restrict__ ssum,
                                      const float* __restrict__ h, float* __restrict__ agg) {
  int w = (blockIdx.x * blockDim.x + threadIdx.x) >> 5;
  int lane = threadIdx.x & 31;
  if (w >= E + N) return;
  int s = (w < E) ? ei[w] : (w - E);
  int d = (w < E) ? ei[E + w] : (w - E);
  float alpha[HEADS_];
#pragma unroll
  for (int hd = 0; hd < HEADS_; ++hd) {
    float v = als[s * HEADS_ + hd] + ald[d * HEADS_ + hd];
    v = v > 0.f ? v : 0.2f * v;
    alpha[hd] = expf(v - key2f(mkey[d * HEADS_ + hd])) / ssum[d * HEADS_ + hd];
  }
  const int c0 = lane * 4;
  const int CH_ = HIDC / HEADS_;
  float a = alpha[c0 / CH_];                        // 4 channels share a head (CH_>=32)
  const float4 hv = ((const float4*)(h + (size_t)s * HIDC))[lane];
  float* o = agg + (size_t)d * HIDC + c0;
  atomicAdd(o + 0, hv.x * a);
  atomicAdd(o + 1, hv.y * a);
  atomicAdd(o + 2, hv.z * a);
  atomicAdd(o + 3, hv.w * a);
}

// ---------------------------------------------------------------- bias + ELU
__global__ void bias_elu_kernel(const float* __restrict__ agg, const float* __restrict__ b,
                                float* __restrict__ out, long long total) {
  long long i = (long long)blockIdx.x * blockDim.x + threadIdx.x;
  if (i >= total) return;
  float v = agg[i] + b[i & (HIDC - 1)];
  out[i] = v > 0.f ? v : expm1f(v);
}

// ---------------------------------------------------------------- pooling
__global__ void pool_accum_kernel(const float* __restrict__ h, const int* __restrict__ batch,
                                  float* __restrict__ psum, float* __restrict__ pcnt, int n) {
  int w = (blockIdx.x * blockDim.x + threadIdx.x) >> 5;
  int lane = threadIdx.x & 31;
  if (w >= n) return;
  int g = batch[w];
  const float4 v = ((const float4*)(h + (size_t)w * HIDC))[lane];
  float* o = psum + (size_t)g * HIDC + lane * 4;
  atomicAdd(o + 0, v.x);
  atomicAdd(o + 1, v.y);
  atomicAdd(o + 2, v.z);
  atomicAdd(o + 3, v.w);
  if (lane == 0) atomicAdd(&pcnt[g], 1.0f);
}

// ---------------------------------------------------------------- classifier + output
// out[0:G] = logits, out[G : G + G*128] = h_pool (tuple flattened in order).
__global__ void classifier_kernel(const float* __restrict__ psum, const float* __restrict__ pcnt,
                                  const float* __restrict__ Wc1, const float* __restrict__ bc1,
                                  const float* __restrict__ Wc2, const float* __restrict__ bc2,
                                  float* __restrict__ out, int G_) {
  int g = blockIdx.x;
  int j = threadIdx.x;  // 64 threads
  __shared__ float hp[HIDC];
  __shared__ float red[64];
  float inv = 1.0f / fmaxf(pcnt[g], 1.0f);
  for (int c = j; c < HIDC; c += 64) {
    float v = psum[(size_t)g * HIDC + c] * inv;
    hp[c] = v;
    out[G_ + (size_t)g * HIDC + c] = v;
  }
  __syncthreads();
  float t = 0.f;
  for (int c = 0; c < HIDC; ++c) t += hp[c] * Wc1[c * 64 + j];  // Wc1 [128,64]
  t = fmaxf(t + bc1[j], 0.f);
  red[j] = t * Wc2[j];                                          // Wc2 [64,1]
  __syncthreads();
  for (int sft = 32; sft > 0; sft >>= 1) {
    if (j < sft) red[j] += red[j + sft];
    __syncthreads();
  }
  if (j == 0) out[g] = red[0] + bc2[0];
}

// ---------------------------------------------------------------- launch
extern "C" void kernel_launch(void* const* d_in, const int* in_sizes, int n_in,
                              void* d_out, int out_size, void* d_ws, size_t ws_size,
                              hipStream_t stream) {
  const int*   x_lex = (const int*)d_in[0];
  const int*   ei    = (const int*)d_in[1];
  const int*   batch = (const int*)d_in[2];
  const float* emb   = (const float*)d_in[3];
  const float* W1    = (const float*)d_in[4];
  const float* a1s   = (const float*)d_in[5];
  const float* a1d   = (const float*)d_in[6];
  const float* b1    = (const float*)d_in[7];
  const float* W2    = (const float*)d_in[8];
  const float* a2s   = (const float*)d_in[9];
  const float* a2d   = (const float*)d_in[10];
  const float* b2    = (const float*)d_in[11];
  const float* Wc1   = (const float*)d_in[12];
  const float* bc1   = (const float*)d_in[13];
  const float* Wc2   = (const float*)d_in[14];
  const float* bc2   = (const float*)d_in[15];

  const int N = in_sizes[0];
  const int E = in_sizes[1] / 2;
  const int G = out_size / (HIDC + 1);
  const int Etot = E + N;

  // workspace: bufA,bufB [N,128] ; als,ald [N,4] ; mkey,ssum [N,4] ; psum [G,128]+cnt
  float* ws = (float*)d_ws;
  float* bufA = ws;
  float* bufB = bufA + (size_t)N * HIDC;
  float* als  = bufB + (size_t)N * HIDC;
  float* ald  = als + (size_t)N * 4;
  unsigned* mkey = (unsigned*)(ald + (size_t)N * 4);
  float* ssum = (float*)(mkey + (size_t)N * 4);
  float* psum = ssum + (size_t)N * 4;
  // pcnt lives right after psum (zeroed together)

  auto cdiv = [](long long a, long long b) { return (int)((a + b - 1) / b); };

  // ---------------- layer 1 (4 heads x 32 ch) ----------------
  gemm16_wmma<<<cdiv(N, 16), 128, 0, stream>>>(emb, x_lex, W1, bufA, N);
  attn_logits_kernel<<<cdiv((long long)N * 4, 256), 256, 0, stream>>>(bufA, a1s, a1d, als, ald, N, 4, 32);
  fill_u32_kernel<<<cdiv((long long)N * 4, 256), 256, 0, stream>>>(mkey, 0u, (long long)N * 4);
  fill_u32_kernel<<<cdiv((long long)N * 4, 256), 256, 0, stream>>>((uint32_t*)ssum, 0u, (long long)N * 4);
  fill_u32_kernel<<<cdiv((long long)N * HIDC, 256), 256, 0, stream>>>((uint32_t*)bufB, 0u, (long long)N * HIDC);
  edge_max_kernel<4><<<cdiv(Etot, 256), 256, 0, stream>>>(ei, E, N, als, ald, mkey);
  edge_expsum_kernel<4><<<cdiv(Etot, 256), 256, 0, stream>>>(ei, E, N, als, ald, mkey, ssum);
  edge_aggregate_kernel<4><<<cdiv((long long)Etot * 32, 256), 256, 0, stream>>>(ei, E, N, als, ald, mkey, ssum, bufA, bufB);
  bias_elu_kernel<<<cdiv((long long)N * HIDC, 256), 256, 0, stream>>>(bufB, b1, bufA, (long long)N * HIDC);

  // ---------------- layer 2 (1 head x 128 ch) ----------------
  gemm16_wmma<<<cdiv(N, 16), 128, 0, stream>>>(bufA, nullptr, W2, bufB, N);
  attn_logits_kernel<<<cdiv(N, 256), 256, 0, stream>>>(bufB, a2s, a2d, als, ald, N, 1, 128);
  fill_u32_kernel<<<cdiv(N, 256), 256, 0, stream>>>(mkey, 0u, N);
  fill_u32_kernel<<<cdiv(N, 256), 256, 0, stream>>>((uint32_t*)ssum, 0u, N);
  fill_u32_kernel<<<cdiv((long long)N * HIDC, 256), 256, 0, stream>>>((uint32_t*)bufA, 0u, (long long)N * HIDC);
  edge_max_kernel<1><<<cdiv(Etot, 256), 256, 0, stream>>>(ei, E, N, als, ald, mkey);
  edge_expsum_kernel<1><<<cdiv(Etot, 256), 256, 0, stream>>>(ei, E, N, als, ald, mkey, ssum);
  edge_aggregate_kernel<1><<<cdiv((long long)Etot * 32, 256), 256, 0, stream>>>(ei, E, N, als, ald, mkey, ssum, bufB, bufA);
  bias_elu_kernel<<<cdiv((long long)N * HIDC, 256), 256, 0, stream>>>(bufA, b2, bufA, (long long)N * HIDC);

  // ---------------- pool + classifier ----------------
  fill_u32_kernel<<<cdiv((long long)G * HIDC + G, 256), 256, 0, stream>>>((uint32_t*)psum, 0u, (long long)G * HIDC + G);
  pool_accum_kernel<<<cdiv((long long)N * 32, 256), 256, 0, stream>>>(bufA, batch, psum, psum + (size_t)G * HIDC, N);
  classifier_kernel<<<G, 64, 0, stream>>>(psum, psum + (size_t)G * HIDC, Wc1, bc1, Wc2, bc2, (float*)d_out, G);
}